// SCTConv_66606352826622
// MI455X (gfx1250) — compile-verified
//
#include <hip/hip_runtime.h>
#include <hip/hip_bf16.h>

typedef __attribute__((ext_vector_type(2))) float v2f;
typedef __attribute__((ext_vector_type(8))) float v8f;

#define NEG_SLOPE 0.01f

// ---------------------------------------------------------------- utilities
__global__ void k_zero(float* __restrict__ p, long n) {
    long i = (long)blockIdx.x * blockDim.x + threadIdx.x;
    if (i < n) p[i] = 0.f;
}

// deg[col[e]] += w[e]
__global__ void k_deg(const int* __restrict__ col, const float* __restrict__ w,
                      float* __restrict__ deg, int E) {
    int e = blockIdx.x * blockDim.x + threadIdx.x;
    if (e < E) atomicAdd(&deg[col[e]], w[e]);
}

// dg = rsqrt(deg+1), di = 1/deg
__global__ void k_scales(const float* __restrict__ deg, float* __restrict__ dg,
                         float* __restrict__ di, int N) {
    int n = blockIdx.x * blockDim.x + threadIdx.x;
    if (n < N) {
        float d = deg[n];
        dg[n] = rsqrtf(d + 1.0f);
        di[n] = 1.0f / d;
    }
}

// Fused diagonal-scaled SpMM: y[row[e]] += w[e] * s[col[e]] * x[col[e]]
// One wave per edge, float4 per lane (32 lanes * 4 = D=128).
// x is read with row stride `xstride` (dense D or h_all channel stride 6*D).
__global__ void k_spmm(float* __restrict__ y, const float* __restrict__ x,
                       long xstride, const float* __restrict__ s,
                       const int* __restrict__ row, const int* __restrict__ col,
                       const float* __restrict__ w, int E) {
    long t = (long)blockIdx.x * blockDim.x + threadIdx.x;
    long e = t >> 5;                 // wave32: edge per wave
    if (e >= E) return;
    int d0 = (int)(t & 31) * 4;
    int c = col[e], r = row[e];
    float we = w[e] * s[c];
    const float4 v = *reinterpret_cast<const float4*>(x + (long)c * xstride + d0);
    float* yp = y + (long)r * 128 + d0;
    atomicAdd(yp + 0, we * v.x);
    atomicAdd(yp + 1, we * v.y);
    atomicAdd(yp + 2, we * v.z);
    atomicAdd(yp + 3, we * v.w);
}

// dst[n*dstride+d] = (y[n*D+d] + x[n*xstride+d]*dg[n]) * dg[n]
__global__ void k_gcn_update(float* __restrict__ dst, long dstride,
                             const float* __restrict__ y, const float* __restrict__ x,
                             long xstride, const float* __restrict__ dg, int N, int D) {
    long i = (long)blockIdx.x * blockDim.x + threadIdx.x;
    long nd = (long)N * D;
    if (i < nd) {
        long n = i / D, d = i % D;
        float sc = dg[n];
        dst[n * dstride + d] = (y[i] + x[n * xstride + d] * sc) * sc;
    }
}

// fpn = 0.5*fpc + 0.5*y ; optionally chan[n*cstride+d] = |fpc - fpn|^m
__global__ void k_sct_update(float* __restrict__ fpn, const float* __restrict__ fpc,
                             const float* __restrict__ y, float* __restrict__ chan,
                             long cstride, const int* __restrict__ mptr, int N, int D) {
    long i = (long)blockIdx.x * blockDim.x + threadIdx.x;
    long nd = (long)N * D;
    if (i < nd) {
        float prev = fpc[i];
        float nv = 0.5f * prev + 0.5f * y[i];
        fpn[i] = nv;
        if (chan) {
            int m = *mptr;
            float p = fabsf(prev - nv);
            float v = (m == 1) ? p : powf(p, (float)m);
            long n = i / D, d = i % D;
            chan[n * cstride + d] = v;
        }
    }
}

// in-place leaky_relu on h_all channels 0..2 (layout [N,6,D])
__global__ void k_leaky3(float* __restrict__ h_all, int N, int D) {
    long i = (long)blockIdx.x * blockDim.x + threadIdx.x;
    long tot = (long)N * 3 * D;
    if (i < tot) {
        long n = i / (3 * D);
        long r = i % (3 * D);
        float* p = h_all + n * 6 * D + r;
        float v = *p;
        *p = v > 0.f ? v : NEG_SLOPE * v;
    }
}

// one wave per node: 6-channel attention + weighted mean -> h_prime
// (the relu(X)@a[:D] term is constant across channels and cancels in softmax)
__global__ void k_attn(float* __restrict__ h_prime, const float* __restrict__ h_all,
                       const float* __restrict__ a, int N, int D) {
    long t = (long)blockIdx.x * blockDim.x + threadIdx.x;
    long node = t >> 5;
    int lane = (int)(t & 31);
    if (node >= N) return;
    const float* hb = h_all + node * 6 * (long)D;
    const float* a2 = a + D;
    float e[6];
#pragma unroll
    for (int c = 0; c < 6; ++c) {
        float s = 0.f;
        for (int d = lane; d < D; d += 32) {
            float hv = hb[c * D + d];
            s += fmaxf(hv, 0.f) * a2[d];
        }
#pragma unroll
        for (int off = 16; off > 0; off >>= 1) s += __shfl_xor(s, off, 32);
        e[c] = s;
    }
    float mx = e[0];
#pragma unroll
    for (int c = 1; c < 6; ++c) mx = fmaxf(mx, e[c]);
    float at[6]; float sum = 0.f;
#pragma unroll
    for (int c = 0; c < 6; ++c) { at[c] = __expf(e[c] - mx); sum += at[c]; }
    float inv = 1.0f / (6.0f * sum);
    for (int d = lane; d < D; d += 32) {
        float acc = 0.f;
#pragma unroll
        for (int c = 0; c < 6; ++c) acc += at[c] * hb[c * D + d];
        h_prime[node * D + d] = acc * inv;
    }
}

// out = leaky(H @ W^T + b).  D=128.  Grid: N/16 blocks x 256 threads.
// Wave w in a block owns the 16x16 output tile (rowTile=blockIdx, colTile=w).
// V_WMMA_F32_16X16X4_F32: A 16x4 f32 (v2f/lane), B 4x16 f32 (v2f/lane), C/D v8f.
__global__ __launch_bounds__(256) void k_gemm_leaky(
    float* __restrict__ out, const float* __restrict__ H,
    const float* __restrict__ W, const float* __restrict__ bias, int N) {
    const int D = 128;
    int colBase = (threadIdx.x >> 5) * 16;          // 8 waves -> 8 col tiles
    int lane = threadIdx.x & 31;
    int rowBase = blockIdx.x * 16;
    int l15 = lane & 15;
    int khalf = (lane >> 4) * 2;                    // lanes 16-31 hold K+2,K+3
    const float* hrow = H + (long)(rowBase + l15) * D + khalf;   // A: M=l15
    const float* wrow = W + (long)(colBase + l15) * D + khalf;   // B: N=l15 (W^T)
    v8f acc = {0.f, 0.f, 0.f, 0.f, 0.f, 0.f, 0.f, 0.f};
    for (int k = 0; k < D; k += 4) {
        v2f av = *reinterpret_cast<const v2f*>(hrow + k);
        v2f bv = *reinterpret_cast<const v2f*>(wrow + k);
        acc = __builtin_amdgcn_wmma_f32_16x16x4_f32(
            /*neg_a=*/false, av, /*neg_b=*/false, bv,
            /*c_mod=*/(short)0, acc, /*reuse_a=*/false, /*reuse_b=*/false);
    }
    int nIdx = colBase + l15;
    float bval = bias[nIdx];
    int rOff = (lane >> 4) * 8;                     // lanes 16-31: M = 8+r
#pragma unroll
    for (int r = 0; r < 8; ++r) {
        int m = rowBase + rOff + r;
        float v = acc[r] + bval;
        out[(long)m * D + nIdx] = v > 0.f ? v : NEG_SLOPE * v;
    }
}

// ---------------------------------------------------------------- launcher
extern "C" void kernel_launch(void* const* d_in, const int* in_sizes, int n_in,
                              void* d_out, int out_size, void* d_ws, size_t ws_size,
                              hipStream_t stream) {
    const int D = 128;
    const int N = in_sizes[0] / D;          // 50000
    const int E = in_sizes[2];              // edge_weight length = 900000

    const float* X  = (const float*)d_in[0];
    const int*   ei = (const int*)d_in[1];  // [2,E]: row then col
    const int*   row = ei;
    const int*   col = ei + E;
    const float* w  = (const float*)d_in[2];
    const float* W1 = (const float*)d_in[3];
    const float* b1 = (const float*)d_in[4];
    const float* W2 = (const float*)d_in[5];
    const float* b2 = (const float*)d_in[6];
    const float* av = (const float*)d_in[7];
    const int*   mp = (const int*)d_in[8];
    float* out = (float*)d_out;

    // workspace layout (floats): deg[N] dg[N] di[N] y[ND] fa[ND] fb[ND] h_all[6*ND]
    float* ws = (float*)d_ws;
    const long ND = (long)N * D;
    float* deg   = ws;
    float* dg    = deg + N;
    float* di    = dg + N;
    float* y     = di + N;
    float* fa    = y + ND;
    float* fb    = fa + ND;
    float* h_all = fb + ND;

    const dim3 blk(256);
    auto nb = [](long n) { return dim3((unsigned)((n + 255) / 256)); };

    // degrees & scales
    k_zero  <<<nb(N), blk, 0, stream>>>(deg, N);
    k_deg   <<<nb(E), blk, 0, stream>>>(col, w, deg, E);
    k_scales<<<nb(N), blk, 0, stream>>>(deg, dg, di, N);

    // 3x GCN propagation -> h_all channels 0..2 (pre-activation stored, relu'd after)
    // x_{t+1} = (spmm(dg*x_t) + dg*x_t) * dg  -- diagonal scale fused into spmm gather
    const float* xcur = X; long xs = D;
    for (int it = 0; it < 3; ++it) {
        k_zero<<<nb(ND), blk, 0, stream>>>(y, ND);
        k_spmm<<<nb((long)E * 32), blk, 0, stream>>>(y, xcur, xs, dg, row, col, w, E);
        float* dst = h_all + it * D;
        k_gcn_update<<<nb(ND), blk, 0, stream>>>(dst, (long)6 * D, y, xcur, xs, dg, N, D);
        xcur = dst; xs = (long)6 * D;
    }
    k_leaky3<<<nb((long)N * 3 * D), blk, 0, stream>>>(h_all, N, D);

    // 4x scattering -> abs-diff^m channels 3..5
    // fp_{t+1} = 0.5*fp_t + 0.5*spmm(di*fp_t)  -- di fused into spmm gather
    const float* fpc = X; float* fpn = fa;
    for (int it = 0; it < 4; ++it) {
        k_zero<<<nb(ND), blk, 0, stream>>>(y, ND);
        k_spmm<<<nb((long)E * 32), blk, 0, stream>>>(y, fpc, D, di, row, col, w, E);
        float* chan = (it >= 1) ? (h_all + (3 + it - 1) * D) : nullptr;
        k_sct_update<<<nb(ND), blk, 0, stream>>>(fpn, fpc, y, chan, (long)6 * D, mp, N, D);
        fpc = fpn; fpn = (fpn == fa) ? fb : fa;
    }
    // after loop: fa holds sct2, fb holds sct3; both free now

    // attention + weighted mean -> h_prime (reuse fa)
    k_attn<<<nb((long)N * 32), blk, 0, stream>>>(fa, h_all, av, N, D);

    // two dense GEMMs with WMMA f32 (N = 3125*16, 128 = 8*16: exact tiling)
    k_gemm_leaky<<<dim3(N / 16), blk, 0, stream>>>(fb, fa, W1, b1, N);
    k_gemm_leaky<<<dim3(N / 16), blk, 0, stream>>>(out, fb, W2, b2, N);
}